// DCNv2Function_47115791237770
// MI455X (gfx1250) — compile-verified
//
#include <hip/hip_runtime.h>

typedef __attribute__((ext_vector_type(2))) float v2f;
typedef __attribute__((ext_vector_type(8))) float v8f;

#define B_   4
#define C_   64
#define H_   96
#define W_   96
#define CO_  64
#define KW_  3
#define K_   9
#define HW_  (H_*W_)        // 9216
#define CKK_ (C_*K_)        // 576
#define PIX_TILE 128
#define TILES_PER_IMG (HW_/PIX_TILE)  // 72

// LDS pitches in float2 units; chosen so the two lane-halves (row delta)
// shift by 32 banks -> conflict-free ds_load_b64 fragment reads.
#define WP_ 80    // weight-pair pitch: 2*WP_ floats = 160 mod 64 = 32
#define SP_ 144   // sample-pair pitch: 2*SP_ floats = 288 mod 64 = 32

// Re-layout weight[co][c][t] -> w2[t][c*64 + co] so the per-tap LDS staging
// load in the main kernel is fully coalesced.
__global__ __launch_bounds__(256)
void dcnv2_prep_weights(const float* __restrict__ w, float* __restrict__ w2) {
  int i = blockIdx.x * 256 + threadIdx.x;
  if (i < K_ * C_ * CO_) {
    int t  = i / (C_ * CO_);
    int r  = i - t * (C_ * CO_);
    int c  = r >> 6;
    int co = r & 63;
    w2[i] = w[co * CKK_ + c * K_ + t];
  }
}

__global__ __launch_bounds__(256)
void dcnv2_wmma_kernel(const float* __restrict__ input,
                       const float* __restrict__ offset,
                       const float* __restrict__ mask,
                       const float* __restrict__ weight,
                       const float* __restrict__ w2,
                       const float* __restrict__ bias,
                       float* __restrict__ out,
                       int use_w2)
{
  __shared__ float lds_w[32 * WP_ * 2];  // 20480 B: W_t pairs [c/2][co]{c even, c odd}
  __shared__ float lds_s[32 * SP_ * 2];  // 36864 B: samples  [c/2][pix]{c even, c odd}

  const int tid  = threadIdx.x;
  const int wg   = blockIdx.x;
  const int b    = wg / TILES_PER_IMG;
  const int pix0 = (wg % TILES_PER_IMG) * PIX_TILE;

  // sampling-phase role: 2 threads per pixel, 32 channels each
  const int pix  = tid & 127;
  const int half = tid >> 7;
  const int p    = pix0 + pix;
  const int y    = p / W_;
  const int x    = p - y * W_;

  const float* off_b = offset + (size_t)b * (2 * K_) * HW_;
  const float* msk_b = mask   + (size_t)b * K_ * HW_;
  const float* img_h = input  + (size_t)b * C_ * HW_ + (size_t)half * 32 * HW_;

  // WMMA-phase role: wave -> (M block, 4 N blocks)
  const int lo   = tid & 15;
  const int hi   = (tid >> 4) & 1;
  const int wave = tid >> 5;
  const int mb   = wave & 3;          // Co rows mb*16 .. mb*16+15
  const int nbb  = (wave >> 2) * 4;   // pixel blocks nbb .. nbb+3

  v8f acc[4];
#pragma unroll
  for (int j = 0; j < 4; ++j)
#pragma unroll
    for (int e = 0; e < 8; ++e) acc[j][e] = 0.0f;

#pragma unroll 1
  for (int t = 0; t < K_; ++t) {
    // ---- stage W_t[c][co] = weight[co][c][t] into LDS as K-pairs over c ----
    {
      const float* wsrc = w2 + t * (C_ * CO_);
#pragma unroll
      for (int it = 0; it < 16; ++it) {
        int i  = tid + it * 256;     // 0..4095
        int c  = i >> 6;
        int co = i & 63;
        float wv = use_w2 ? wsrc[i] : weight[co * CKK_ + c * K_ + t];
        lds_w[((c >> 1) * WP_ + co) * 2 + (c & 1)] = wv;
      }
    }

    // ---- per-pixel bilinear metadata (validity & mask folded into weights) ----
    float dy = off_b[(2 * t)     * HW_ + p];
    float dx = off_b[(2 * t + 1) * HW_ + p];
    float mv = msk_b[t * HW_ + p];
    float py = (float)(y - 1 + t / KW_) + dy;
    float px = (float)(x - 1 + t % KW_) + dx;
    float fy = floorf(py), fx = floorf(px);
    float wy = py - fy,    wx = px - fx;
    int iy0 = (int)fy, ix0 = (int)fx;
    int iy1 = iy0 + 1, ix1 = ix0 + 1;
    int vy0 = (iy0 >= 0) && (iy0 < H_);
    int vy1 = (iy1 >= 0) && (iy1 < H_);
    int vx0 = (ix0 >= 0) && (ix0 < W_);
    int vx1 = (ix1 >= 0) && (ix1 < W_);
    int cy0 = iy0 < 0 ? 0 : (iy0 > H_ - 1 ? H_ - 1 : iy0);
    int cy1 = iy1 < 0 ? 0 : (iy1 > H_ - 1 ? H_ - 1 : iy1);
    int cx0 = ix0 < 0 ? 0 : (ix0 > W_ - 1 ? W_ - 1 : ix0);
    int cx1 = ix1 < 0 ? 0 : (ix1 > W_ - 1 ? W_ - 1 : ix1);
    int i00 = cy0 * W_ + cx0, i01 = cy0 * W_ + cx1;
    int i10 = cy1 * W_ + cx0, i11 = cy1 * W_ + cx1;
    float w00 = (vy0 && vx0) ? (1.f - wy) * (1.f - wx) * mv : 0.f;
    float w01 = (vy0 && vx1) ? (1.f - wy) * wx * mv         : 0.f;
    float w10 = (vy1 && vx0) ? wy * (1.f - wx) * mv         : 0.f;
    float w11 = (vy1 && vx1) ? wy * wx * mv                 : 0.f;

    // ---- gather-sample 32 channels for this pixel into LDS ----
    const float* ic = img_h;
#pragma unroll 4
    for (int cc = 0; cc < 32; ++cc) {
      float v = w00 * ic[i00] + w01 * ic[i01] + w10 * ic[i10] + w11 * ic[i11];
      int c = half * 32 + cc;
      lds_s[((c >> 1) * SP_ + pix) * 2 + (c & 1)] = v;
      ic += HW_;
    }
    __syncthreads();

    // ---- WMMA: D[co][pix] += sum_c W_t[c][co-row] * samp[c][pix] ----
    // fp32 16x16x4 fragment layout: lane%16 = M/N column, lane/16 selects the
    // K-pair, the 2 VGPRs hold K and K+1 -> one ds_load_b64 per fragment.
    const v2f* wp = (const v2f*)lds_w;
    const v2f* sp = (const v2f*)lds_s;
#pragma unroll
    for (int cb = 0; cb < 16; ++cb) {          // 4 channels per step
      v2f a = wp[(2 * cb + hi) * WP_ + mb * 16 + lo];
#pragma unroll
      for (int j = 0; j < 4; ++j) {
        v2f bf = sp[(2 * cb + hi) * SP_ + (nbb + j) * 16 + lo];
        acc[j] = __builtin_amdgcn_wmma_f32_16x16x4_f32(
            false, a, false, bf, (short)0, acc[j], false, false);
      }
    }
    __syncthreads();
  }

  // ---- epilogue: bias + store (16 contiguous floats per half-wave row) ----
  float bi[8];
#pragma unroll
  for (int i = 0; i < 8; ++i) bi[i] = bias[mb * 16 + 8 * hi + i];
#pragma unroll
  for (int j = 0; j < 4; ++j) {
    int ncol = pix0 + (nbb + j) * 16 + lo;
#pragma unroll
    for (int i = 0; i < 8; ++i) {
      int co = mb * 16 + 8 * hi + i;   // C/D layout: VGPR i, lane-half -> M = i + 8*hi
      out[((size_t)b * CO_ + co) * HW_ + ncol] = acc[j][i] + bi[i];
    }
  }
}

extern "C" void kernel_launch(void* const* d_in, const int* in_sizes, int n_in,
                              void* d_out, int out_size, void* d_ws, size_t ws_size,
                              hipStream_t stream) {
  (void)in_sizes; (void)n_in; (void)out_size;
  const float* input  = (const float*)d_in[0];
  const float* offset = (const float*)d_in[1];
  const float* mask   = (const float*)d_in[2];
  const float* weight = (const float*)d_in[3];
  const float* bias   = (const float*)d_in[4];
  float* out = (float*)d_out;
  float* w2  = (float*)d_ws;

  const size_t w2_bytes = (size_t)K_ * C_ * CO_ * sizeof(float);
  const int use_w2 = (d_ws != nullptr && ws_size >= w2_bytes) ? 1 : 0;

  if (use_w2) {
    int n = K_ * C_ * CO_;  // 36864
    dcnv2_prep_weights<<<(n + 255) / 256, 256, 0, stream>>>(weight, w2);
  }
  dcnv2_wmma_kernel<<<B_ * TILES_PER_IMG, 256, 0, stream>>>(
      input, offset, mask, weight, w2, bias, out, use_w2);
}